// SSCMBlock_14388140441603
// MI455X (gfx1250) — compile-verified
//
#include <hip/hip_runtime.h>

typedef __attribute__((ext_vector_type(2))) float v2f;
typedef __attribute__((ext_vector_type(8))) float v8f;

#define B_   8
#define C_   128
#define H_   32
#define W_   32
#define L_   1024
#define DI_  256
#define NST_ 16
#define DTR_ 8
#define P_   (B_*L_)   // 8192 pixel-rows

__device__ __forceinline__ float silu_f(float x) { return x / (1.f + __expf(-x)); }

// sequence-index -> row-major pixel index for the 4 spatial scan directions
__device__ __forceinline__ int perm_idx(int dir, int t) {
    int u = (dir & 1) ? (L_ - 1 - t) : t;
    if (dir >= 2) u = ((u & 31) << 5) | (u >> 5);   // column-major traversal (H=W=32)
    return u;
}

// async global->LDS 16-byte copy (CDNA5 TDM-lite path, tracked by ASYNCcnt).
// Low 32 bits of a generic LDS pointer are the LDS byte offset (ISA 10.2).
__device__ __forceinline__ void async_ld16(void* lds_dst, const float* gsrc) {
    unsigned l = (unsigned)(unsigned long long)lds_dst;
    asm volatile("global_load_async_to_lds_b128 %0, %1, off"
                 :: "v"(l), "v"(gsrc)
                 : "memory");
}
__device__ __forceinline__ void wait_async0() {
    asm volatile("s_wait_asynccnt 0x0" ::: "memory");
}

// ---------------------------------------------------------------------------
// Generic GEMM:  C[M x Npad] = alpha * A[M x K] * W[Nreal x K]^T   (fp32 WMMA)
// 256 threads (8 waves), tile 64x64, K-chunks of 16.
// Double-buffered async global->LDS staging, ONE barrier per chunk.
// ---------------------------------------------------------------------------
__global__ __launch_bounds__(256)
void gemm_nt_wmma(const float* __restrict__ A, const float* __restrict__ Wt,
                  float* __restrict__ Cout, int M, int Nreal, int Npad, int K,
                  int lda, int ldb, int ldc, float alpha)
{
    __shared__ float sA[2][64 * 20];
    __shared__ float sB[2][64 * 20];
    const int tid  = threadIdx.x;
    const int lane = tid & 31;
    const int wave = tid >> 5;
    const int wm   = wave >> 1;       // 0..3 (row groups of 16)
    const int wn   = wave & 1;        // 0..1 (col groups of 32)
    const int m16  = lane & 15;
    const int hi   = lane >> 4;
    const int rowBase = blockIdx.y * 64;
    const int colBase = blockIdx.x * 64;

    const int r  = tid >> 2;          // 0..63 : staged row
    const int c4 = (tid & 3) << 2;    // 0,4,8,12 : staged col (float4)
    const bool bvalid = (colBase + r) < Nreal;

    // zero-fill padded B rows once (they are never async-written)
    if (!bvalid) {
        float4 z = make_float4(0.f, 0.f, 0.f, 0.f);
        *(float4*)&sB[0][r * 20 + c4] = z;
        *(float4*)&sB[1][r * 20 + c4] = z;
    }

    const float* gA = A  + (size_t)(rowBase + r) * lda + c4;
    const float* gB = Wt + (size_t)(colBase + r) * ldb + c4;

    // prologue: issue tile 0
    async_ld16(&sA[0][r * 20 + c4], gA);
    if (bvalid) async_ld16(&sB[0][r * 20 + c4], gB);

    v8f acc0 = {}; v8f acc1 = {};
    const int nk = K >> 4;

    for (int i = 0; i < nk; ++i) {
        const int buf = i & 1;
        wait_async0();                 // my tile-i copies have landed in LDS
        __syncthreads();               // everyone's tile i ready; buf^1 free
        if (i + 1 < nk) {              // overlap next tile's DMA with compute
            async_ld16(&sA[buf ^ 1][r * 20 + c4], gA + (size_t)(i + 1) * 16);
            if (bvalid) async_ld16(&sB[buf ^ 1][r * 20 + c4], gB + (size_t)(i + 1) * 16);
        }
#pragma unroll
        for (int kk = 0; kk < 16; kk += 4) {
            const int ka = kk + 2 * hi;
            v2f a, b0, b1;
            a.x = sA[buf][(wm * 16 + m16) * 20 + ka];
            a.y = sA[buf][(wm * 16 + m16) * 20 + ka + 1];
            int n0 = wn * 32 + m16;
            b0.x = sB[buf][n0 * 20 + ka];        b0.y = sB[buf][n0 * 20 + ka + 1];
            b1.x = sB[buf][(n0 + 16) * 20 + ka]; b1.y = sB[buf][(n0 + 16) * 20 + ka + 1];
            acc0 = __builtin_amdgcn_wmma_f32_16x16x4_f32(false, a, false, b0,
                                                         (short)0, acc0, false, false);
            acc1 = __builtin_amdgcn_wmma_f32_16x16x4_f32(false, a, false, b1,
                                                         (short)0, acc1, false, false);
        }
    }
#pragma unroll
    for (int rr = 0; rr < 8; ++rr) {
        int row  = rowBase + wm * 16 + rr + 8 * hi;
        int col0 = colBase + wn * 32 + m16;
        if (col0 < Npad)      Cout[(size_t)row * ldc + col0]      = alpha * acc0[rr];
        if (col0 + 16 < Npad) Cout[(size_t)row * ldc + col0 + 16] = alpha * acc1[rr];
    }
}

// ---------------------------------------------------------------------------
// 3x3 conv as implicit-im2col WMMA GEMM: M=8192 pixels, N=128, K=1152
// X layout (B,C,H,W); W layout (Co, Ci*9); out (pixel-major) [P_ x 128]
// A: scalar im2col gather; B (weights): async global->LDS.
// ---------------------------------------------------------------------------
__device__ __forceinline__ const float* im2col_addr(const float* X, int p, int k, bool& ok)
{
    int ci = k / 9;
    int rem = k - ci * 9;
    int kh = rem / 3, kw = rem - kh * 3;
    int b = p >> 10, l = p & 1023;
    int iy = (l >> 5) + kh - 1, ix = (l & 31) + kw - 1;
    ok = ((unsigned)iy < 32u) && ((unsigned)ix < 32u);
    return X + (size_t)((b * C_ + ci) * H_ + iy) * W_ + ix;
}

__global__ __launch_bounds__(256)
void conv3x3_wmma(const float* __restrict__ X, const float* __restrict__ Wt,
                  float* __restrict__ Cout)
{
    __shared__ float sA[64 * 20];
    __shared__ float sB[64 * 20];
    const int tid  = threadIdx.x;
    const int lane = tid & 31;
    const int wave = tid >> 5;
    const int wm   = wave >> 1;
    const int wn   = wave & 1;
    const int m16  = lane & 15;
    const int hi   = lane >> 4;
    const int rowBase = blockIdx.y * 64;
    const int colBase = blockIdx.x * 64;
    const int K = C_ * 9;

    const int r  = tid >> 2;
    const int c4 = (tid & 3) << 2;
    const float* gB = Wt + (size_t)(colBase + r) * K + c4;

    v8f acc0 = {}; v8f acc1 = {};

    for (int k0 = 0; k0 < K; k0 += 16) {
        // weights: async copy straight into LDS
        async_ld16(&sB[r * 20 + c4], gB + k0);
        // activations: im2col gather
#pragma unroll
        for (int i = 0; i < 4; ++i) {
            int e = (tid << 2) + i;
            int rr = e >> 4, c = e & 15;
            bool ok;
            const float* src = im2col_addr(X, rowBase + rr, k0 + c, ok);
            sA[rr * 20 + c] = ok ? *src : 0.f;
        }
        if (k0 + 16 < K) {             // near-cache prefetch of next im2col chunk
            bool ok;
            const float* nsrc = im2col_addr(X, rowBase + (tid >> 2), k0 + 16 + (tid & 15), ok);
            if (ok) __builtin_prefetch(nsrc, 0, 3);
        }
        wait_async0();
        __syncthreads();
#pragma unroll
        for (int kk = 0; kk < 16; kk += 4) {
            const int ka = kk + 2 * hi;
            v2f a, b0, b1;
            a.x = sA[(wm * 16 + m16) * 20 + ka];
            a.y = sA[(wm * 16 + m16) * 20 + ka + 1];
            int n0 = wn * 32 + m16;
            b0.x = sB[n0 * 20 + ka];        b0.y = sB[n0 * 20 + ka + 1];
            b1.x = sB[(n0 + 16) * 20 + ka]; b1.y = sB[(n0 + 16) * 20 + ka + 1];
            acc0 = __builtin_amdgcn_wmma_f32_16x16x4_f32(false, a, false, b0,
                                                         (short)0, acc0, false, false);
            acc1 = __builtin_amdgcn_wmma_f32_16x16x4_f32(false, a, false, b1,
                                                         (short)0, acc1, false, false);
        }
        __syncthreads();
    }
#pragma unroll
    for (int rr = 0; rr < 8; ++rr) {
        int row  = rowBase + wm * 16 + rr + 8 * hi;
        int col0 = colBase + wn * 32 + m16;
        Cout[(size_t)row * C_ + col0]      = acc0[rr];
        Cout[(size_t)row * C_ + col0 + 16] = acc1[rr];
    }
}

// ---------------------------------------------------------------------------
// BatchNorm (training-mode batch stats) + ReLU ; also transposes to (B,C,H,W)
// ---------------------------------------------------------------------------
__global__ __launch_bounds__(256)
void bn_stats(const float* __restrict__ CT, float* __restrict__ stats)
{
    int c = blockIdx.x;
    __shared__ float ss[256], sq[256];
    float s = 0.f, q = 0.f;
    for (int p = threadIdx.x; p < P_; p += 256) {
        float v = CT[(size_t)p * C_ + c];
        s += v; q += v * v;
    }
    ss[threadIdx.x] = s; sq[threadIdx.x] = q;
    __syncthreads();
    for (int o = 128; o > 0; o >>= 1) {
        if (threadIdx.x < (unsigned)o) {
            ss[threadIdx.x] += ss[threadIdx.x + o];
            sq[threadIdx.x] += sq[threadIdx.x + o];
        }
        __syncthreads();
    }
    if (threadIdx.x == 0) {
        float m = ss[0] / (float)P_;
        stats[c]      = m;
        stats[C_ + c] = sq[0] / (float)P_ - m * m;
    }
}

__global__ void bn_apply_relu(const float* __restrict__ CT, const float* __restrict__ stats,
                              const float* __restrict__ g, const float* __restrict__ bb,
                              float* __restrict__ OUT)
{
    int tid = blockIdx.x * 256 + threadIdx.x;       // 1M, (p, c) with c fast
    int c = tid & 127, p = tid >> 7;
    float m = stats[c], v = stats[C_ + c];
    float y = (CT[tid] - m) * rsqrtf(v + 1e-5f) * g[c] + bb[c];
    y = fmaxf(y, 0.f);
    int b = p >> 10, l = p & 1023;
    OUT[(size_t)(b * C_ + c) * L_ + l] = y;         // (B,C,H,W)
}

__global__ void build_xf(const float* __restrict__ X, float* __restrict__ XF)
{
    int tid = blockIdx.x * 256 + threadIdx.x;       // 1M, (p, c) c fast
    int c = tid & 127, p = tid >> 7;
    int b = p >> 10, l = p & 1023;
    XF[tid] = X[(size_t)(b * C_ + c) * L_ + l];
}

// ---------------------------------------------------------------------------
// Mamba pieces
// ---------------------------------------------------------------------------
__global__ void dwconv_silu(const float* __restrict__ XZ, const float* __restrict__ cw,
                            const float* __restrict__ cb, float* __restrict__ XC, int dir)
{
    int tid = blockIdx.x * 256 + threadIdx.x;       // 2M
    int ch = tid & 255, pt = tid >> 8;
    int b = pt >> 10, t = pt & 1023;
    float v = cb[ch];
#pragma unroll
    for (int k = 0; k < 4; ++k) {
        int tt = t - 3 + k;
        if (tt >= 0) {
            int l = perm_idx(dir, tt);
            v += cw[ch * 4 + k] * XZ[(size_t)(b * L_ + l) * 512 + ch];
        }
    }
    XC[(size_t)pt * DI_ + ch] = silu_f(v);
}

__global__ void spe_xc(const float* __restrict__ XZ, const float* __restrict__ cw,
                       const float* __restrict__ cb, float* __restrict__ XC)
{
    int tid = blockIdx.x * 256 + threadIdx.x;       // 2M ; L=1 -> only tap k=3
    int ch = tid & 255, p = tid >> 8;
    float v = XZ[(size_t)p * 512 + ch] * cw[ch * 4 + 3] + cb[ch];
    XC[tid] = silu_f(v);
}

__global__ void dtproj_softplus(const float* __restrict__ DBL, const float* __restrict__ dtw,
                                const float* __restrict__ dtb, float* __restrict__ DT)
{
    int tid = blockIdx.x * 256 + threadIdx.x;       // 2M
    int ch = tid & 255, p = tid >> 8;
    float v = dtb[ch];
#pragma unroll
    for (int r = 0; r < DTR_; ++r) v += DBL[(size_t)p * 48 + r] * dtw[ch * DTR_ + r];
    DT[tid] = (v > 20.f) ? v : __logf(1.f + __expf(v));
}

// sequential SSM scan: 16 state lanes per (b,d) chain; 2 chains per wave32
__global__ __launch_bounds__(256)
void ssm_scan_spa(const float* __restrict__ DT, const float* __restrict__ XC,
                  const float* __restrict__ DBL, const float* __restrict__ Alog,
                  const float* __restrict__ Dp, float* __restrict__ Y)
{
    int wave  = blockIdx.x * 8 + (threadIdx.x >> 5);
    int lane  = threadIdx.x & 31;
    int chain = wave * 2 + (lane >> 4);
    int n = lane & 15;
    int b = chain >> 8, d = chain & 255;
    float Ad = -__expf(Alog[d * NST_ + n]);
    float Dd = Dp[d];
    float h = 0.f;
    for (int t = 0; t < L_; ++t) {
        size_t base = (size_t)(b * L_ + t);
        float dtv = DT[base * DI_ + d];
        float xv  = XC[base * DI_ + d];
        float Bv  = DBL[base * 48 + DTR_ + n];
        float Cv  = DBL[base * 48 + DTR_ + NST_ + n];
        h = __expf(dtv * Ad) * h + (dtv * xv) * Bv;
        float part = h * Cv;
        part += __shfl_xor(part, 8, 32);
        part += __shfl_xor(part, 4, 32);
        part += __shfl_xor(part, 2, 32);
        part += __shfl_xor(part, 1, 32);
        if (n == 0) Y[base * DI_ + d] = part + xv * Dd;
    }
}

__global__ void combine_spa(const float* __restrict__ Y, const float* __restrict__ XZ,
                            float* __restrict__ G, int dir)
{
    int tid = blockIdx.x * 256 + threadIdx.x;       // 2M
    int ch = tid & 255, pt = tid >> 8;
    int b = pt >> 10, t = pt & 1023;
    int l = perm_idx(dir, t);
    float z = XZ[(size_t)(b * L_ + l) * 512 + 256 + ch];
    G[tid] += Y[tid] * silu_f(z);
}

__global__ void spe_bc(const float* __restrict__ DBL, float* __restrict__ BCv)
{
    int p = blockIdx.x * 256 + threadIdx.x;
    if (p >= P_) return;
    float s = 0.f;
#pragma unroll
    for (int n = 0; n < NST_; ++n)
        s += DBL[(size_t)p * 48 + DTR_ + n] * DBL[(size_t)p * 48 + DTR_ + NST_ + n];
    BCv[p] = s;
}

__global__ void combine_spe(const float* __restrict__ DT, const float* __restrict__ XC,
                            const float* __restrict__ BCv, const float* __restrict__ Dp,
                            const float* __restrict__ XZ, float* __restrict__ G)
{
    int tid = blockIdx.x * 256 + threadIdx.x;       // 2M
    int ch = tid & 255, p = tid >> 8;
    float xc = XC[tid];
    float y  = DT[tid] * xc * BCv[p] + xc * Dp[ch];
    G[tid] = y * silu_f(XZ[(size_t)p * 512 + 256 + ch]);
}

// ---------------------------------------------------------------------------
// Attention gates + final combine
// ---------------------------------------------------------------------------
__global__ void spa_pool_k(const float* __restrict__ FL, float* __restrict__ PM,
                           float* __restrict__ PX)
{
    int p = blockIdx.x * 256 + threadIdx.x;
    if (p >= P_) return;
    int b = p >> 10, l = p & 1023;
    float m = 0.f, mx = -3.402823466e38f;
    for (int c = 0; c < C_; ++c) {
        float v = FL[(size_t)(b * C_ + c) * L_ + l];
        m += v; mx = fmaxf(mx, v);
    }
    PM[p] = m / (float)C_; PX[p] = mx;
}

__global__ __launch_bounds__(32)
void spa_att_k(const float* __restrict__ PM, const float* __restrict__ PX,
               const float* __restrict__ w, const float* __restrict__ bias,
               float* __restrict__ ASPA)
{
    int by = blockIdx.x;            // b*32 + y
    int b = by >> 5, y = by & 31;
    int x = threadIdx.x;
    float acc = bias[0];
#pragma unroll
    for (int kh = 0; kh < 7; ++kh) {
        int iy = y + kh - 3;
        if ((unsigned)iy >= 32u) continue;
#pragma unroll
        for (int kw = 0; kw < 7; ++kw) {
            int ix = x + kw - 3;
            if ((unsigned)ix >= 32u) continue;
            int pp = (b << 10) + (iy << 5) + ix;
            acc += PM[pp] * w[kh * 7 + kw] + PX[pp] * w[49 + kh * 7 + kw];
        }
    }
    float mx = acc;                  // softmax across W=32 (one wave)
    for (int o = 16; o > 0; o >>= 1) mx = fmaxf(mx, __shfl_xor(mx, o, 32));
    float e = __expf(acc - mx);
    float s = e;
    for (int o = 16; o > 0; o >>= 1) s += __shfl_xor(s, o, 32);
    ASPA[(b << 10) + (y << 5) + x] = e / s;
}

__global__ void spe_pool_k(const float* __restrict__ FL, float* __restrict__ PM2,
                           float* __restrict__ PX2)
{
    int bc = blockIdx.x * 256 + threadIdx.x;
    if (bc >= B_ * C_) return;
    float m = 0.f, mx = -3.402823466e38f;
    for (int l = 0; l < L_; ++l) {
        float v = FL[(size_t)bc * L_ + l];
        m += v; mx = fmaxf(mx, v);
    }
    PM2[bc] = m / (float)L_; PX2[bc] = mx;
}

__global__ void spe_att_k(const float* __restrict__ PM2, const float* __restrict__ PX2,
                          const float* __restrict__ w, const float* __restrict__ bias,
                          float* __restrict__ ASPE)
{
    int bc = blockIdx.x * 256 + threadIdx.x;
    if (bc >= B_ * C_) return;
    int b = bc >> 7, c = bc & 127;
    float acc = bias[c];
    for (int c2 = 0; c2 < C_; ++c2)
        acc += w[c * 2 * C_ + c2] * PM2[b * C_ + c2] + w[c * 2 * C_ + C_ + c2] * PX2[b * C_ + c2];
    ASPE[bc] = 1.f / (1.f + __expf(-acc));
}

__global__ void final_k(const float* __restrict__ FSPA, const float* __restrict__ FSPE,
                        const float* __restrict__ FL, const float* __restrict__ ASPA,
                        const float* __restrict__ ASPE, float* __restrict__ OUT)
{
    int tid = blockIdx.x * 256 + threadIdx.x;       // 1M, (b,c,l)
    int l = tid & 1023, bc = tid >> 10;
    int b = bc >> 7, c = bc & 127;
    size_t p = (size_t)(b << 10) + l;
    float fl = FL[tid];
    OUT[tid] = (FSPA[p * C_ + c] + fl) * ASPA[(b << 10) + l]
             + (FSPE[p * C_ + c] + fl) * ASPE[bc];
}

// ---------------------------------------------------------------------------
extern "C" void kernel_launch(void* const* d_in, const int* in_sizes, int n_in,
                              void* d_out, int out_size, void* d_ws, size_t ws_size,
                              hipStream_t stream)
{
    (void)in_sizes; (void)n_in; (void)out_size; (void)ws_size;
    const float* x         = (const float*)d_in[0];
    const float* conv1_w   = (const float*)d_in[1];
    const float* bn1_g     = (const float*)d_in[2];
    const float* bn1_b     = (const float*)d_in[3];
    const float* conv2_w   = (const float*)d_in[4];
    const float* bn2_g     = (const float*)d_in[5];
    const float* bn2_b     = (const float*)d_in[6];
    const float* spa_att_w = (const float*)d_in[7];
    const float* spa_att_b = (const float*)d_in[8];
    const float* spe_att_w = (const float*)d_in[9];
    const float* spe_att_b = (const float*)d_in[10];
    const float* spa[9]; for (int i = 0; i < 9; ++i) spa[i] = (const float*)d_in[11 + i];
    const float* spe[9]; for (int i = 0; i < 9; ++i) spe[i] = (const float*)d_in[20 + i];

    float* ws = (float*)d_ws;
    size_t o = 0;
    float* FLOC = ws + o; o += (size_t)P_ * C_;
    float* CTMP = ws + o; o += (size_t)P_ * C_;
    float* STAT = ws + o; o += 512;
    float* XF   = ws + o; o += (size_t)P_ * C_;
    float* XZ   = ws + o; o += (size_t)P_ * 512;
    float* XC   = ws + o; o += (size_t)P_ * DI_;
    float* DBL  = ws + o; o += (size_t)P_ * 48;
    float* DT   = ws + o; o += (size_t)P_ * DI_;
    float* Yb   = ws + o; o += (size_t)P_ * DI_;
    float* G    = ws + o; o += (size_t)P_ * DI_;
    float* FSPA = ws + o; o += (size_t)P_ * C_;
    float* FSPE = ws + o; o += (size_t)P_ * C_;
    float* PM   = ws + o; o += P_;
    float* PX   = ws + o; o += P_;
    float* ASPA = ws + o; o += P_;
    float* PM2  = ws + o; o += 1024;
    float* PX2  = ws + o; o += 1024;
    float* ASPE = ws + o; o += 1024;
    float* BCv  = ws + o; o += P_;

    dim3 blk(256);

    // ---- ConvBlock ----
    build_xf<<<4096, blk, 0, stream>>>(x, XF);
    conv3x3_wmma<<<dim3(2, 128), blk, 0, stream>>>(x, conv1_w, CTMP);
    bn_stats<<<128, blk, 0, stream>>>(CTMP, STAT);
    bn_apply_relu<<<4096, blk, 0, stream>>>(CTMP, STAT, bn1_g, bn1_b, FLOC);
    conv3x3_wmma<<<dim3(2, 128), blk, 0, stream>>>(FLOC, conv2_w, CTMP);
    bn_stats<<<128, blk, 0, stream>>>(CTMP, STAT);
    bn_apply_relu<<<4096, blk, 0, stream>>>(CTMP, STAT, bn2_g, bn2_b, FLOC);

    // ---- Spatial Mamba: shared in-projection, 4 directional scans ----
    gemm_nt_wmma<<<dim3(8, 128), blk, 0, stream>>>(XF, spa[0], XZ,
        P_, 512, 512, C_, C_, C_, 512, 1.f);
    hipMemsetAsync(G, 0, (size_t)P_ * DI_ * sizeof(float), stream);
    for (int dir = 0; dir < 4; ++dir) {
        dwconv_silu<<<8192, blk, 0, stream>>>(XZ, spa[1], spa[2], XC, dir);
        gemm_nt_wmma<<<dim3(1, 128), blk, 0, stream>>>(XC, spa[3], DBL,
            P_, 40, 48, DI_, DI_, DI_, 48, 1.f);
        dtproj_softplus<<<8192, blk, 0, stream>>>(DBL, spa[4], spa[5], DT);
        ssm_scan_spa<<<128, blk, 0, stream>>>(DT, XC, DBL, spa[6], spa[7], Yb);
        combine_spa<<<8192, blk, 0, stream>>>(Yb, XZ, G, dir);
    }
    gemm_nt_wmma<<<dim3(2, 128), blk, 0, stream>>>(G, spa[8], FSPA,
        P_, 128, 128, DI_, DI_, DI_, 128, 0.25f);

    // ---- Spectral Mamba (L=1 closed form) ----
    gemm_nt_wmma<<<dim3(8, 128), blk, 0, stream>>>(XF, spe[0], XZ,
        P_, 512, 512, C_, C_, C_, 512, 1.f);
    spe_xc<<<8192, blk, 0, stream>>>(XZ, spe[1], spe[2], XC);
    gemm_nt_wmma<<<dim3(1, 128), blk, 0, stream>>>(XC, spe[3], DBL,
        P_, 40, 48, DI_, DI_, DI_, 48, 1.f);
    dtproj_softplus<<<8192, blk, 0, stream>>>(DBL, spe[4], spe[5], DT);
    spe_bc<<<32, blk, 0, stream>>>(DBL, BCv);
    combine_spe<<<8192, blk, 0, stream>>>(DT, XC, BCv, spe[7], XZ, G);
    gemm_nt_wmma<<<dim3(2, 128), blk, 0, stream>>>(G, spe[8], FSPE,
        P_, 128, 128, DI_, DI_, DI_, 128, 1.f);

    // ---- Attention gates + output ----
    spa_pool_k<<<32, blk, 0, stream>>>(FLOC, PM, PX);
    spa_att_k<<<256, dim3(32), 0, stream>>>(PM, PX, spa_att_w, spa_att_b, ASPA);
    spe_pool_k<<<4, blk, 0, stream>>>(FLOC, PM2, PX2);
    spe_att_k<<<4, blk, 0, stream>>>(PM2, PX2, spe_att_w, spe_att_b, ASPE);
    final_k<<<4096, blk, 0, stream>>>(FSPA, FSPE, FLOC, ASPA, ASPE, (float*)d_out);
}